// MultiVariateGaussian_12343736009111
// MI455X (gfx1250) — compile-verified
//
#include <hip/hip_runtime.h>
#include <hip/hip_bf16.h>

// MultiVariateGaussian: mean + per-position inverse covariance via Woodbury.
//   cov_i = lam*I + W^T W,  W = centered_i / sqrt(B-1)  (W is 32x128)
//   cov_i^{-1} = (1/lam) * (I - W^T S^{-1} W),  S = lam*I_32 + W W^T (32x32)
// Pipeline:
//   K1 (prep): coalesced read of (B,C,HW); computes mean (output #1) and writes
//              centered/scaled W to d_ws in (HW,B,C) layout (51MB, L2-resident).
//   K2 (woodbury): one wave per position; async global->LDS staging of W;
//              all GEMMs via V_WMMA_F32_16X16X4_F32; NT stores for the 207MB
//              inverse-covariance stream.

#define B_  32
#define C_  128
#define HW_ 3136
#define LAMBDA_ (0.01f + 1e-5f)

#define ITILE 32
#define CTILE 32
#define PITCH 33   // LDS row pitch (floats) -> conflict-free il-major and cl-major access

typedef float v2f __attribute__((ext_vector_type(2)));
typedef float v8f __attribute__((ext_vector_type(8)));

// ---- CDNA5 async global->LDS copies (ASYNCcnt-tracked) ----
__device__ __forceinline__ unsigned lds_off(const void* p) {
    const char* pc = (const char*)p;
    auto p3 = (__attribute__((address_space(3))) const char*)pc;
    return (unsigned)(unsigned long long)p3;
}
__device__ __forceinline__ void async_g2l_b32(const float* g, const float* l) {
    unsigned lo = lds_off(l);
    asm volatile("global_load_async_to_lds_b32 %0, %1, off"
                 :: "v"(lo), "v"(g) : "memory");
}
__device__ __forceinline__ void async_g2l_b128(const float* g, const float* l) {
    unsigned lo = lds_off(l);
    asm volatile("global_load_async_to_lds_b128 %0, %1, off"
                 :: "v"(lo), "v"(g) : "memory");
}
__device__ __forceinline__ void wait_async0() {
    asm volatile("s_wait_asynccnt 0" ::: "memory");
}

// ---- WMMA f32 16x16x4 wrapper ----
__device__ __forceinline__ v8f wmma_f32(v2f a, v2f b, v8f c) {
    return __builtin_amdgcn_wmma_f32_16x16x4_f32(
        false, a, false, b, (short)0, c, false, false);
}

// ---- Fragment loaders from row-major LDS matrices ----
__device__ __forceinline__ v2f loadA(const float* M, int ld, int m0, int k0, int lane) {
    int r  = lane & 15;
    int kk = k0 + ((lane >> 4) << 1);
    v2f a; a.x = M[(m0 + r) * ld + kk]; a.y = M[(m0 + r) * ld + kk + 1];
    return a;
}
__device__ __forceinline__ v2f loadAT(const float* M, int ld, int m0, int k0, int lane) {
    int r  = lane & 15;
    int kk = k0 + ((lane >> 4) << 1);
    v2f a; a.x = M[kk * ld + (m0 + r)]; a.y = M[(kk + 1) * ld + (m0 + r)];
    return a;
}
__device__ __forceinline__ v2f loadB(const float* M, int ld, int k0, int n0, int lane) {
    int n  = n0 + (lane & 15);
    int kk = k0 + ((lane >> 4) << 1);
    v2f b; b.x = M[kk * ld + n]; b.y = M[(kk + 1) * ld + n];
    return b;
}
__device__ __forceinline__ v2f loadBT(const float* M, int ld, int k0, int n0, int lane) {
    int n  = n0 + (lane & 15);
    int kk = k0 + ((lane >> 4) << 1);
    v2f b; b.x = M[n * ld + kk]; b.y = M[n * ld + kk + 1];
    return b;
}

// ---- Kernel 1: fused mean + center + transpose ----
// Grid: (HW/32, C/32). Block: 256 threads. Each block stages a 32i x 32c x 32b
// raw tile in LDS (async copies, coalesced 128B/wave), reduces the mean over b,
// then writes centered/scaled W transposed to (HW,B,C), coalesced along c.
__global__ __launch_bounds__(256) void mvg_prep_kernel(const float* __restrict__ emb,
                                                       float* __restrict__ mean,
                                                       float* __restrict__ Wt) {
    extern __shared__ float sm[];
    float* raw = sm;                       // [b][cl][il], pitch PITCH
    float* mu  = sm + B_ * (CTILE * PITCH); // [cl][il], pitch PITCH

    const int i0  = blockIdx.x * ITILE;
    const int c0  = blockIdx.y * CTILE;
    const int tid = threadIdx.x;
    const int il  = tid & 31;
    const int c8  = tid >> 5;              // wave-uniform: 0..7

    // Stage: async global->LDS, 128B contiguous per wave-instruction.
    for (int b = 0; b < B_; ++b)
        for (int cg = 0; cg < 4; ++cg) {
            int cl = cg * 8 + c8;
            const float* g = emb + ((size_t)(b * C_ + c0 + cl)) * HW_ + i0 + il;
            async_g2l_b32(g, &raw[b * (CTILE * PITCH) + cl * PITCH + il]);
        }
    wait_async0();
    __syncthreads();

    // Mean over b (bank-conflict free: lanes vary il).
    for (int cg = 0; cg < 4; ++cg) {
        int cl = cg * 8 + c8;
        float s = 0.f;
#pragma unroll
        for (int b = 0; b < B_; ++b)
            s += raw[b * (CTILE * PITCH) + cl * PITCH + il];
        float m = s * (1.0f / B_);
        mu[cl * PITCH + il] = m;
        mean[(size_t)(c0 + cl) * HW_ + i0 + il] = m;   // coalesced along il
    }
    __syncthreads();

    // Transposed centered write: lanes vary cl -> coalesced along c in Wt;
    // LDS reads stride PITCH=33 across lanes -> conflict-free.
    const float scale = rsqrtf((float)(B_ - 1));
    const int cl2 = tid & 31;
    const int i8  = tid >> 5;
    for (int b = 0; b < B_; ++b)
        for (int ig = 0; ig < 4; ++ig) {
            int il2 = ig * 8 + i8;
            float w = (raw[b * (CTILE * PITCH) + cl2 * PITCH + il2]
                       - mu[cl2 * PITCH + il2]) * scale;
            Wt[((size_t)(i0 + il2) * B_ + b) * C_ + c0 + cl2] = w;
        }
}

// ---- Kernel 2: Woodbury inverse, one wave per position ----
__global__ __launch_bounds__(32) void mvg_woodbury_kernel(const float* __restrict__ Wt,
                                                          float* __restrict__ invcov) {
    __shared__ __align__(16) float Wl[B_ * C_];  // 32x128 row-major [b][c]
    __shared__ float Sm[B_ * B_];
    __shared__ float Si[B_ * B_];
    __shared__ float Pt[16 * B_];

    const int pos  = blockIdx.x;
    const int lane = threadIdx.x;

    // Stage W slice: contiguous 16KB, async b128 (512B per wave-instruction).
    const float* wsrc = Wt + (size_t)pos * (B_ * C_);
    for (int t = lane * 4; t < B_ * C_; t += 32 * 4)
        async_g2l_b128(wsrc + t, &Wl[t]);
    wait_async0();
    __syncthreads();

    // ---- S = lam*I + W W^T (32x32), K=128 ----
#pragma unroll
    for (int tm = 0; tm < 2; ++tm) {
#pragma unroll
        for (int tn = 0; tn < 2; ++tn) {
            v8f acc = {};
            for (int k0 = 0; k0 < C_; k0 += 4) {
                v2f a = loadA (Wl, C_, tm * 16, k0, lane);
                v2f b = loadBT(Wl, C_, k0, tn * 16, lane);   // B = W^T
                acc = wmma_f32(a, b, acc);
            }
            int n     = tn * 16 + (lane & 15);
            int mbase = tm * 16 + ((lane >> 4) << 3);
#pragma unroll
            for (int r = 0; r < 8; ++r) {
                int m = mbase + r;
                Sm[m * B_ + n] = acc[r] + ((m == n) ? LAMBDA_ : 0.f);
                Si[m * B_ + n] = (m == n) ? 1.f : 0.f;
            }
        }
    }
    __syncthreads();

    // ---- Gauss-Jordan inversion of S (SPD; no pivoting). Lane = column. ----
    for (int k = 0; k < B_; ++k) {
        float rp  = 1.0f / Sm[k * B_ + k];
        float skj = Sm[k * B_ + lane] * rp;
        float ikj = Si[k * B_ + lane] * rp;
        Sm[k * B_ + lane] = skj;
        Si[k * B_ + lane] = ikj;
        __syncthreads();
        for (int i = 0; i < B_; ++i) {
            if (i == k) continue;
            float f = Sm[i * B_ + k];           // uniform LDS read (broadcast)
            Sm[i * B_ + lane] -= f * skj;
            Si[i * B_ + lane] -= f * ikj;
        }
        __syncthreads();
    }

    // ---- inv = (1/lam)*(I - W^T S^{-1} W), streamed out with NT stores ----
    const float inv_lam = 1.0f / LAMBDA_;
    float* outp = invcov + (size_t)pos * (C_ * C_);

    for (int c1 = 0; c1 < 8; ++c1) {
        // Pt (16x32) = (W^T rows c1*16..) * S^{-1},  K = 32
#pragma unroll
        for (int tn = 0; tn < 2; ++tn) {
            v8f acc = {};
#pragma unroll
            for (int k0 = 0; k0 < B_; k0 += 4) {
                v2f a = loadAT(Wl, C_, c1 * 16, k0, lane);
                v2f b = loadB (Si, B_, k0, tn * 16, lane);
                acc = wmma_f32(a, b, acc);
            }
            int n     = tn * 16 + (lane & 15);
            int mbase = (lane >> 4) << 3;
#pragma unroll
            for (int r = 0; r < 8; ++r)
                Pt[(mbase + r) * B_ + n] = acc[r];
        }
        __syncthreads();

        // Z row-block = Pt (16x32) * W (32x128), emit final tiles.
        for (int c2 = 0; c2 < 8; ++c2) {
            v8f acc = {};
#pragma unroll
            for (int k0 = 0; k0 < B_; k0 += 4) {
                v2f a = loadA(Pt, B_, 0, k0, lane);
                v2f b = loadB(Wl, C_, k0, c2 * 16, lane);
                acc = wmma_f32(a, b, acc);
            }
            int n     = c2 * 16 + (lane & 15);
            int mbase = c1 * 16 + ((lane >> 4) << 3);
#pragma unroll
            for (int r = 0; r < 8; ++r) {
                int m = mbase + r;
                float v = inv_lam * (((m == n) ? 1.f : 0.f) - acc[r]);
                __builtin_nontemporal_store(v, &outp[(size_t)m * C_ + n]);
            }
        }
        __syncthreads();   // before Pt is overwritten by next c1
    }
}

extern "C" void kernel_launch(void* const* d_in, const int* in_sizes, int n_in,
                              void* d_out, int out_size, void* d_ws, size_t ws_size,
                              hipStream_t stream) {
    (void)in_sizes; (void)n_in; (void)ws_size; (void)out_size;
    const float* emb = (const float*)d_in[0];
    float* out    = (float*)d_out;
    float* mean   = out;                        // (C, HW) = 401408 floats
    float* invcov = out + (size_t)C_ * HW_;     // (HW, C, C)
    float* Wt     = (float*)d_ws;               // (HW, B, C) centered/scaled, ~51MB

    dim3 gridA(HW_ / ITILE, C_ / CTILE);        // (98, 4)
    size_t smA = (size_t)(B_ * CTILE * PITCH + CTILE * PITCH) * sizeof(float); // ~136KB
    mvg_prep_kernel<<<gridA, 256, smA, stream>>>(emb, mean, Wt);
    mvg_woodbury_kernel<<<HW_, 32, 0, stream>>>(Wt, invcov);
}